// TStokenizer_10290741641415
// MI455X (gfx1250) — compile-verified
//
#include <hip/hip_runtime.h>
#include <hip/hip_bf16.h>

// ---------------- constants ----------------
#define Bn   64
#define Ln   5000
#define Fn   12
#define Hn   64
#define NEn  1024
#define WLn  32
#define Tn   156            // L // WL
#define KSn  3
#define NBn  8
#define ROWS (Bn*Ln)        // 320000
#define LOOKBACK 16         // max (KS-1)*dil = 2*8
#define DEC_ELEMS (Bn*Ln*Fn)   // 3,840,000
#define LN_EPS 1e-8f
#define XSTR 66             // padded sX row stride (even -> b64 aligned; bank 2m+k)
#define OSTR 65             // padded sOut row stride (bank m+c, conflict-free)

typedef float v2f __attribute__((ext_vector_type(2)));
typedef float v8f __attribute__((ext_vector_type(8)));

__device__ __forceinline__ v8f wmma4(v2f a, v2f b, v8f c) {
  // D = A(16x4,f32) * B(4x16,f32) + C(16x16,f32)
  return __builtin_amdgcn_wmma_f32_16x16x4_f32(false, a, false, b, (short)0, c,
                                               false, false);
}

__device__ __forceinline__ v8f v8zero() {
  v8f z = {0.f,0.f,0.f,0.f,0.f,0.f,0.f,0.f};
  return z;
}

__device__ __forceinline__ v2f ldsv2(const float* p) {
  return *reinterpret_cast<const v2f*>(p);
}

// ---------------- encoder input linear: (B*L,12) @ (12,64) + b ----------------
__global__ void lin_in_kernel(const float* __restrict__ inp,
                              const float* __restrict__ W,
                              const float* __restrict__ bias,
                              float* __restrict__ out) {
  int idx = blockIdx.x * 256 + threadIdx.x;        // over B*L*H
  if (idx >= ROWS * Hn) return;
  int c  = idx & 63;
  int bl = idx >> 6;
  float acc = bias[c];
  const float* xr = inp + bl * Fn;
#pragma unroll
  for (int f = 0; f < Fn; ++f) acc += xr[f] * W[f * Hn + c];
  out[idx] = acc;
}

// ---------------- fused causal dilated conv (K=3) + LN + ReLU (+ residual) ----
// out_pre[b,l,co] = bias[co] + sum_{i,ci} x[b, l-(2-i)*dil, ci] * w[i,ci,co]
// h = relu(LN(out_pre));  out = useRes ? alpha*h + res : h
__global__ void conv_ln_kernel(const float* __restrict__ xin,
                               const float* __restrict__ res,
                               float* __restrict__ out,
                               const float* __restrict__ w,     // (KS,H,H)
                               const float* __restrict__ bias,  // (H)
                               const float* __restrict__ gamma, // (H)
                               const float* __restrict__ beta,  // (H)
                               const float* __restrict__ alphaPtr,
                               int dil, int useRes) {
  // Weights packed per K-step: sWp[(tap*16 + k/4)*256 + n*4 + (k&3)]
  //  -> B fragment (k..k+1, n) is one aligned b64; banks (4n+kp)%64 conflict-free.
  __shared__ float sWp[KSn * 16 * 256];             // 12288 floats, 48 KB
  __shared__ float sX[(LOOKBACK + 64) * XSTR];      // 5280 floats
  __shared__ float sOut[64 * OSTR];                 // 4160 floats
  __shared__ float sBias[Hn], sG[Hn], sBe[Hn];
  __shared__ float sS[128], sS2[128];

  const int tid = threadIdx.x;                      // 128 threads = 4 waves
  const int l0  = blockIdx.x * 64;
  const int b   = blockIdx.y;

  // ---- stage packed weights + padded input window + params ----
  for (int i = tid; i < KSn * Hn * Hn; i += 128) {
    int tap = i >> 12;            // /4096
    int r   = i & 4095;
    int k   = r >> 6;
    int n   = r & 63;
    sWp[(tap * 16 + (k >> 2)) * 256 + n * 4 + (k & 3)] = w[i];
  }
  for (int i = tid; i < (LOOKBACK + 64) * Hn; i += 128) {
    int row = i >> 6;
    int c   = i & 63;
    int l   = l0 - LOOKBACK + row;
    sX[row * XSTR + c] = (l >= 0 && l < Ln) ? xin[(b * Ln + l) * Hn + c] : 0.f;
  }
  if (tid < Hn) { sBias[tid] = bias[tid]; sG[tid] = gamma[tid]; sBe[tid] = beta[tid]; }
  __syncthreads();

  // ---- WMMA: each wave computes 16 rows x 64 channels ----
  const int lane  = tid & 31;
  const int wv    = tid >> 5;
  const int m0    = wv * 16;
  const int mrow  = lane & 15;
  const int kp    = (lane >> 4) << 1;               // 0 (lanes 0-15) or 2 (16-31)

  v8f acc[4];
  acc[0] = v8zero(); acc[1] = v8zero(); acc[2] = v8zero(); acc[3] = v8zero();

  for (int tap = 0; tap < KSn; ++tap) {
    const int roff = (2 - tap) * dil;
    const float* aBase = &sX[(LOOKBACK + m0 + mrow - roff) * XSTR + kp];
    const float* wBase = &sWp[tap * 16 * 256 + mrow * 4 + kp];
    for (int cb4 = 0; cb4 < 16; ++cb4) {
      v2f a = ldsv2(aBase + cb4 * 4);
      const float* wk = wBase + cb4 * 256;
#pragma unroll
      for (int nt = 0; nt < 4; ++nt)
        acc[nt] = wmma4(a, ldsv2(wk + nt * 64), acc[nt]);
    }
  }

  // ---- spill accumulators (C layout: vgpr r -> M=r(+8), lane%16 -> N) ----
  const int mhalf = (lane >> 4) * 8;
#pragma unroll
  for (int nt = 0; nt < 4; ++nt)
#pragma unroll
    for (int r = 0; r < 8; ++r)
      sOut[(m0 + mhalf + r) * OSTR + nt * 16 + mrow] = acc[nt][r];
  __syncthreads();

  // ---- LayerNorm + ReLU (+ residual): 2 threads per row, 32 ch each ----
  const int m     = tid & 63;
  const int halfc = (tid >> 6) * 32;
  const int l     = l0 + m;
  float s = 0.f, s2 = 0.f;
  if (l < Ln) {
    for (int c = halfc; c < halfc + 32; ++c) {
      float v = sOut[m * OSTR + c] + sBias[c];
      s += v; s2 += v * v;
    }
  }
  sS[tid] = s; sS2[tid] = s2;
  __syncthreads();
  if (l < Ln) {
    const float st  = sS[m] + sS[m + 64];
    const float st2 = sS2[m] + sS2[m + 64];
    const float mean = st * (1.f / Hn);
    const float var  = st2 * (1.f / Hn) - mean * mean;
    const float inv  = rsqrtf(var + LN_EPS);
    const float al   = useRes ? alphaPtr[0] : 0.f;
    const int base = (b * Ln + l) * Hn;
    for (int c = halfc; c < halfc + 32; ++c) {
      float v = sOut[m * OSTR + c] + sBias[c];
      float h = fmaxf((v - mean) * inv * sG[c] + sBe[c], 0.f);
      out[base + c] = useRes ? (al * h + res[base + c]) : h;
    }
  }
}

// ---------------- generic (rows,64) @ (64,64) + bias, WMMA ----------------
__global__ void linear64_kernel(const float* __restrict__ xin,
                                const float* __restrict__ W,   // (64,64)
                                const float* __restrict__ bias,
                                float* __restrict__ out, int nrows) {
  __shared__ float sWp[16 * 256];                   // packed per K-step
  __shared__ float sB[Hn];
  const int tid = threadIdx.x;
  const int r0  = blockIdx.x * 64;
  for (int i = tid; i < Hn * Hn; i += 128) {
    int k = i >> 6, n = i & 63;
    sWp[(k >> 2) * 256 + n * 4 + (k & 3)] = W[i];
  }
  if (tid < Hn) sB[tid] = bias[tid];
  __syncthreads();

  const int lane = tid & 31, wv = tid >> 5;
  const int m0 = wv * 16, mrow = lane & 15, kp = (lane >> 4) << 1;
  int row = r0 + m0 + mrow;
  if (row >= nrows) row = nrows - 1;                // compute garbage, skip store
  const float* aBase = xin + row * Hn + kp;
  const float* wBase = &sWp[mrow * 4 + kp];

  v8f acc[4];
  acc[0] = v8zero(); acc[1] = v8zero(); acc[2] = v8zero(); acc[3] = v8zero();
  for (int cb4 = 0; cb4 < 16; ++cb4) {
    v2f a = *(const v2f*)(aBase + cb4 * 4);
    const float* wk = wBase + cb4 * 256;
#pragma unroll
    for (int nt = 0; nt < 4; ++nt)
      acc[nt] = wmma4(a, ldsv2(wk + nt * 64), acc[nt]);
  }
  const int mhalf = (lane >> 4) * 8;
#pragma unroll
  for (int nt = 0; nt < 4; ++nt)
#pragma unroll
    for (int r = 0; r < 8; ++r) {
      int orow = r0 + m0 + mhalf + r;
      if (orow < nrows)
        out[orow * Hn + nt * 16 + mrow] = acc[nt][r] + sB[nt * 16 + mrow];
    }
}

// ---------------- patchify GEMM: qi (9984,64) = patches(9984,2048) @ Wt ------
// A(row=(b,t), k=p*64+q) = enc[(b*L + t*WL)*H + k]   (contiguous over k)
// B(k, n) = qin_w[n*2048 + k]  (contiguous v2f per fragment)
__global__ void qin_kernel(const float* __restrict__ enc,
                           const float* __restrict__ qw,   // (H, WL, H)
                           const float* __restrict__ qb,   // (H)
                           float* __restrict__ qi) {
  const int tid = threadIdx.x;
  const int r0  = blockIdx.x * 64;               // 156 blocks * 64 = 9984 rows
  const int lane = tid & 31, wv = tid >> 5;
  const int m0 = wv * 16, mrow = lane & 15, kp = (lane >> 4) << 1;
  const int row = r0 + m0 + mrow;
  const int bi = row / Tn, ti = row % Tn;
  const float* aBase = enc + (bi * Ln + ti * WLn) * Hn + kp;

  v8f acc[4];
  acc[0] = v8zero(); acc[1] = v8zero(); acc[2] = v8zero(); acc[3] = v8zero();
  for (int cb = 0; cb < WLn * Hn; cb += 4) {
    v2f a = *(const v2f*)(aBase + cb);
#pragma unroll
    for (int nt = 0; nt < 4; ++nt) {
      const int n = nt * 16 + mrow;
      v2f bb = *(const v2f*)(qw + n * (WLn * Hn) + cb + kp);
      acc[nt] = wmma4(a, bb, acc[nt]);
    }
  }
  const int mhalf = (lane >> 4) * 8;
#pragma unroll
  for (int nt = 0; nt < 4; ++nt)
#pragma unroll
    for (int r = 0; r < 8; ++r) {
      int orow = r0 + m0 + mhalf + r;            // always < 9984
      qi[orow * Hn + nt * 16 + mrow] = acc[nt][r] + qb[nt * 16 + mrow];
    }
}

// ---------------- codebook column norms ----------------
__global__ void enorm_kernel(const float* __restrict__ embed, float* __restrict__ en) {
  int j = blockIdx.x * 128 + threadIdx.x;        // 8*128 = 1024
  float s = 0.f;
#pragma unroll
  for (int h = 0; h < Hn; ++h) { float e = embed[h * NEn + j]; s += e * e; }
  en[j] = s;
}

__global__ void zero_kernel(float* p) { if (threadIdx.x == 0 && blockIdx.x == 0) p[0] = 0.f; }

// ---------------- VQ: argmin over 1024 codes + gather + diff ----------------
__global__ void vq_kernel(const float* __restrict__ qi,
                          const float* __restrict__ embed,   // (H, NE)
                          const float* __restrict__ enorm,
                          float* __restrict__ q,
                          int* __restrict__ indOut,
                          float* __restrict__ diffOut) {
  __shared__ float sq[Hn];
  __shared__ float sval[128];
  __shared__ int   sidx[128];
  __shared__ int   sbest;
  const int row = blockIdx.x;
  const int tid = threadIdx.x;
  if (tid < Hn) sq[tid] = qi[row * Hn + tid];
  __syncthreads();

  float best = 3.0e38f; int bi = 0;
  for (int j = tid; j < NEn; j += 128) {
    float s = 0.f;
#pragma unroll
    for (int h = 0; h < Hn; ++h) s += sq[h] * embed[h * NEn + j];
    float d = enorm[j] - 2.f * s;                // ||q||^2 constant: drop for argmin
    if (d < best) { best = d; bi = j; }
  }
  sval[tid] = best; sidx[tid] = bi;
  __syncthreads();
  for (int off = 64; off > 0; off >>= 1) {
    if (tid < off) {
      float v2 = sval[tid + off]; int i2 = sidx[tid + off];
      if (v2 < sval[tid] || (v2 == sval[tid] && i2 < sidx[tid])) {
        sval[tid] = v2; sidx[tid] = i2;
      }
    }
    __syncthreads();
  }
  if (tid == 0) { sbest = sidx[0]; indOut[row] = sidx[0]; }
  __syncthreads();

  const int code = sbest;
  float dl = 0.f;
  if (tid < Hn) {
    float e = embed[tid * NEn + code];
    q[row * Hn + tid] = e;
    float df = e - sq[tid];
    dl = df * df;
  }
  sval[tid] = dl;
  __syncthreads();
  for (int off = 64; off > 0; off >>= 1) {
    if (tid < off) sval[tid] += sval[tid + off];
    __syncthreads();
  }
  // diff = mean((q-qi)^2) + BETA*mean = 1.25 * sum / (B*T*H)
  if (tid == 0) atomicAdd(diffOut, sval[0] * (1.25f / ((float)Bn * Tn * Hn)));
}

// ---------------- token upsample: up[b,l,d] = sum_t qout_w[l,t]*q[b,t,d] ------
// K = 156 = 39 k-steps of 4, exact (no tail guards). Per-batch q tile packed
// per K-step into LDS: sQp[(t/4)*256 + n*4 + (t&3)] -> conflict-free b64 frags.
__global__ void qout_kernel(const float* __restrict__ q,      // (B,T,H)
                            const float* __restrict__ Wlt,    // (L,T)
                            const float* __restrict__ bl,     // (L)
                            float* __restrict__ up) {         // (B,L,H)
  __shared__ float sQp[39 * 256];                // 9984 floats, exact
  const int tid = threadIdx.x;
  const int l0  = blockIdx.x * 64;
  const int b   = blockIdx.y;
  for (int i = tid; i < Tn * Hn; i += 128) {
    int t = i >> 6, n = i & 63;
    sQp[(t >> 2) * 256 + n * 4 + (t & 3)] = q[b * Tn * Hn + i];
  }
  __syncthreads();

  const int lane = tid & 31, wv = tid >> 5;
  const int m0 = wv * 16, mrow = lane & 15, kp = (lane >> 4) << 1;
  int l = l0 + m0 + mrow;
  if (l >= Ln) l = Ln - 1;
  const float* aBase = Wlt + l * Tn + kp;        // (l*156+kp)*4 bytes, 8B aligned
  const float* qBase = &sQp[mrow * 4 + kp];

  v8f acc[4];
  acc[0] = v8zero(); acc[1] = v8zero(); acc[2] = v8zero(); acc[3] = v8zero();
  for (int cb4 = 0; cb4 < 39; ++cb4) {
    v2f a = *(const v2f*)(aBase + cb4 * 4);
    const float* qk = qBase + cb4 * 256;
#pragma unroll
    for (int nt = 0; nt < 4; ++nt)
      acc[nt] = wmma4(a, ldsv2(qk + nt * 64), acc[nt]);
  }
  const int mhalf = (lane >> 4) * 8;
#pragma unroll
  for (int nt = 0; nt < 4; ++nt)
#pragma unroll
    for (int r = 0; r < 8; ++r) {
      int ol = l0 + m0 + mhalf + r;
      if (ol < Ln)
        up[(b * Ln + ol) * Hn + nt * 16 + mrow] = acc[nt][r] + bl[ol];
    }
}

// ---------------- final projection: (rows,64) @ (64,12) + b ----------------
__global__ void proj_kernel(const float* __restrict__ y,
                            const float* __restrict__ W,   // (64,12)
                            const float* __restrict__ bias,
                            float* __restrict__ out, int nrows) {
  int idx = blockIdx.x * 256 + threadIdx.x;
  if (idx >= nrows * Fn) return;
  int r = idx / Fn, f = idx % Fn;
  float acc = bias[f];
  const float* yr = y + r * Hn;
#pragma unroll
  for (int c = 0; c < Hn; ++c) acc += yr[c] * W[c * Fn + f];
  out[idx] = acc;
}

// =============================== launch ===============================
extern "C" void kernel_launch(void* const* d_in, const int* in_sizes, int n_in,
                              void* d_out, int out_size, void* d_ws, size_t ws_size,
                              hipStream_t stream) {
  const float* input     = (const float*)d_in[0];
  const float* enc_in_w  = (const float*)d_in[1];
  const float* enc_in_b  = (const float*)d_in[2];
  const float* enc_c1w   = (const float*)d_in[3];
  const float* enc_c1b   = (const float*)d_in[4];
  const float* enc_g1    = (const float*)d_in[5];
  const float* enc_be1   = (const float*)d_in[6];
  const float* enc_c2w   = (const float*)d_in[7];
  const float* enc_c2b   = (const float*)d_in[8];
  const float* enc_g2    = (const float*)d_in[9];
  const float* enc_be2   = (const float*)d_in[10];
  const float* enc_a     = (const float*)d_in[11];
  const float* enc_out_w = (const float*)d_in[12];
  const float* enc_out_b = (const float*)d_in[13];
  const float* qin_w     = (const float*)d_in[14];
  const float* qin_b     = (const float*)d_in[15];
  const float* embed     = (const float*)d_in[16];
  const float* qout_w    = (const float*)d_in[17];
  const float* qout_b    = (const float*)d_in[18];
  const float* dec_c1w   = (const float*)d_in[19];
  const float* dec_c1b   = (const float*)d_in[20];
  const float* dec_g1    = (const float*)d_in[21];
  const float* dec_be1   = (const float*)d_in[22];
  const float* dec_c2w   = (const float*)d_in[23];
  const float* dec_c2b   = (const float*)d_in[24];
  const float* dec_g2    = (const float*)d_in[25];
  const float* dec_be2   = (const float*)d_in[26];
  const float* dec_a     = (const float*)d_in[27];
  const float* dec_out_w = (const float*)d_in[28];
  const float* dec_out_b = (const float*)d_in[29];
  const float* dec_proj_w= (const float*)d_in[30];
  const float* dec_proj_b= (const float*)d_in[31];

  // workspace layout (floats)
  float* X   = (float*)d_ws;                 // (B,L,H)  20,480,000
  float* Hb  = X  + (size_t)ROWS * Hn;       // (B,L,H)  20,480,000
  float* QI  = Hb + (size_t)ROWS * Hn;       // (B,T,H)     638,976
  float* Q   = QI + (size_t)Bn * Tn * Hn;    // (B,T,H)     638,976
  float* EN  = Q  + (size_t)Bn * Tn * Hn;    // (NE)          1,024

  float* outF  = (float*)d_out;              // dec: 3,840,000 floats
  float* diffP = outF + DEC_ELEMS;           // diff scalar
  int*   indP  = (int*)d_out + DEC_ELEMS + 1;// ind: 9984 int32

  const dim3 convGrid(79, Bn);
  const int dils[NBn] = {1, 4, 1, 4, 1, 4, 1, 4};

  // -------- encoder --------
  lin_in_kernel<<<(ROWS * Hn + 255) / 256, 256, 0, stream>>>(input, enc_in_w, enc_in_b, X);
  for (int i = 0; i < NBn; ++i) {
    conv_ln_kernel<<<convGrid, 128, 0, stream>>>(
        X, nullptr, Hb,
        enc_c1w + (size_t)i * KSn * Hn * Hn, enc_c1b + i * Hn,
        enc_g1 + i * Hn, enc_be1 + i * Hn, enc_a + i, dils[i], 0);
    conv_ln_kernel<<<convGrid, 128, 0, stream>>>(
        Hb, X, X,
        enc_c2w + (size_t)i * KSn * Hn * Hn, enc_c2b + i * Hn,
        enc_g2 + i * Hn, enc_be2 + i * Hn, enc_a + i, 2 * dils[i], 1);
  }
  linear64_kernel<<<(ROWS + 63) / 64, 128, 0, stream>>>(X, enc_out_w, enc_out_b, Hb, ROWS);

  // -------- quantize --------
  qin_kernel<<<Bn * Tn / 64, 128, 0, stream>>>(Hb, qin_w, qin_b, QI);
  enorm_kernel<<<NEn / 128, 128, 0, stream>>>(embed, EN);
  zero_kernel<<<1, 1, 0, stream>>>(diffP);
  vq_kernel<<<Bn * Tn, 128, 0, stream>>>(QI, embed, EN, Q, indP, diffP);
  qout_kernel<<<convGrid, 128, 0, stream>>>(Q, qout_w, qout_b, X);

  // -------- decoder --------
  for (int i = 0; i < NBn; ++i) {
    conv_ln_kernel<<<convGrid, 128, 0, stream>>>(
        X, nullptr, Hb,
        dec_c1w + (size_t)i * KSn * Hn * Hn, dec_c1b + i * Hn,
        dec_g1 + i * Hn, dec_be1 + i * Hn, dec_a + i, dils[i], 0);
    conv_ln_kernel<<<convGrid, 128, 0, stream>>>(
        Hb, X, X,
        dec_c2w + (size_t)i * KSn * Hn * Hn, dec_c2b + i * Hn,
        dec_g2 + i * Hn, dec_be2 + i * Hn, dec_a + i, 2 * dils[i], 1);
  }
  linear64_kernel<<<(ROWS + 63) / 64, 128, 0, stream>>>(X, dec_out_w, dec_out_b, Hb, ROWS);
  proj_kernel<<<(ROWS * Fn + 255) / 256, 256, 0, stream>>>(Hb, dec_proj_w, dec_proj_b, outF, ROWS);
}